// TransitionDown_24120536334933
// MI455X (gfx1250) — compile-verified
//
#include <hip/hip_runtime.h>

#define NPTS 32768
#define INC  64
#define OUTC 128
#define MPTS 8192
#define KNN  16

typedef __attribute__((ext_vector_type(2))) float v2f;
typedef __attribute__((ext_vector_type(4))) float v4f;
typedef __attribute__((ext_vector_type(8))) float v8f;
typedef __attribute__((ext_vector_type(4))) unsigned int v4u;
typedef __attribute__((ext_vector_type(8))) int v8i;
typedef __attribute__((ext_vector_type(4))) int v4i;

// ---------------------------------------------------------------------------
// 0) zero the BN accumulators (sum[128], sumsq[128], scale[128], shift[128])
// ---------------------------------------------------------------------------
__global__ __launch_bounds__(512) void init_stats_kernel(float* __restrict__ stats) {
  stats[threadIdx.x] = 0.0f;
}

// ---------------------------------------------------------------------------
// 1) Farthest point sampling: single block, 1024 threads, 32 points/thread in
//    registers. Two-level argmax (wave shfl + LDS across 32 waves).
// ---------------------------------------------------------------------------
__global__ __launch_bounds__(1024) void fps_kernel(const float* __restrict__ pos,
                                                   int* __restrict__ idc) {
  const int tid  = threadIdx.x;
  const int lane = tid & 31;
  const int wave = tid >> 5;
  __shared__ float redD[32];
  __shared__ int   redI[32];
  __shared__ float cur[3];

  float px[32], py[32], pz[32], md[32];
  const float q0x = pos[0], q0y = pos[1], q0z = pos[2];
#pragma unroll
  for (int i = 0; i < 32; ++i) {
    const int p = tid + i * 1024;
    const float X = pos[p * 3 + 0], Y = pos[p * 3 + 1], Z = pos[p * 3 + 2];
    px[i] = X; py[i] = Y; pz[i] = Z;
    const float dx = X - q0x, dy = Y - q0y, dz = Z - q0z;
    md[i] = dx * dx + dy * dy + dz * dz;
  }
  if (tid == 0) idc[0] = 0;

  for (int it = 1; it < MPTS; ++it) {
    float bd = -1.0f; int bi = 0;
#pragma unroll
    for (int i = 0; i < 32; ++i) {
      if (md[i] > bd) { bd = md[i]; bi = tid + i * 1024; }   // lowest index on tie
    }
#pragma unroll
    for (int off = 16; off > 0; off >>= 1) {
      const float od = __shfl_down(bd, off, 32);
      const int   oi = __shfl_down(bi, off, 32);
      if (od > bd || (od == bd && oi < bi)) { bd = od; bi = oi; }
    }
    if (lane == 0) { redD[wave] = bd; redI[wave] = bi; }
    __syncthreads();
    if (wave == 0) {
      bd = redD[lane]; bi = redI[lane];
#pragma unroll
      for (int off = 16; off > 0; off >>= 1) {
        const float od = __shfl_down(bd, off, 32);
        const int   oi = __shfl_down(bi, off, 32);
        if (od > bd || (od == bd && oi < bi)) { bd = od; bi = oi; }
      }
      if (lane == 0) {
        idc[it] = bi;
        cur[0] = pos[bi * 3 + 0];
        cur[1] = pos[bi * 3 + 1];
        cur[2] = pos[bi * 3 + 2];
      }
    }
    __syncthreads();
    const float cx = cur[0], cy = cur[1], cz = cur[2];
#pragma unroll
    for (int i = 0; i < 32; ++i) {
      const float dx = px[i] - cx, dy = py[i] - cy, dz = pz[i] - cz;
      md[i] = fminf(md[i], dx * dx + dy * dy + dz * dz);
    }
  }
}

// ---------------------------------------------------------------------------
// 2) gather sub_pos / sub_batch into the output buffer (raw-bit copy of batch)
// ---------------------------------------------------------------------------
__global__ __launch_bounds__(256) void subgather_kernel(const float* __restrict__ pos,
                                                        const int* __restrict__ batch,
                                                        const int* __restrict__ idc,
                                                        float* __restrict__ out) {
  const int mm = blockIdx.x * 256 + threadIdx.x;
  if (mm >= MPTS) return;
  const int id = idc[mm];
  float* sp = out + (size_t)MPTS * OUTC;
  sp[mm * 3 + 0] = pos[id * 3 + 0];
  sp[mm * 3 + 1] = pos[id * 3 + 1];
  sp[mm * 3 + 2] = pos[id * 3 + 2];
  ((int*)out)[(size_t)MPTS * OUTC + (size_t)MPTS * 3 + mm] = batch[id];
}

// ---------------------------------------------------------------------------
// 3) h = x @ W + b via V_WMMA_F32_16X16X4_F32. W (32KB) staged into LDS with a
//    single Tensor-Data-Mover DMA (TENSOR_LOAD_TO_LDS) issued by wave 0, then
//    s_wait_tensorcnt + block barrier. Each wave: one 16-row stripe across all
//    8 column tiles; A fragments preloaded for the whole K=64.
// ---------------------------------------------------------------------------
__global__ __launch_bounds__(256) void gemm_wmma_kernel(const float* __restrict__ x,
                                                        const float* __restrict__ W,
                                                        const float* __restrict__ bias,
                                                        float* __restrict__ h) {
  __shared__ float sW[INC * OUTC];
  const int tid = threadIdx.x;
  const int wave = tid >> 5, lane = tid & 31;

  if (wave == 0) {                       // uniform branch: only wave 0 issues TDM
    const unsigned long long ga = (unsigned long long)(uintptr_t)W;
    const unsigned int lds = (unsigned int)(uintptr_t)&sW[0];   // LDS byte offset
    v4u g0;
    g0[0] = 1u;                                           // count=1, user mode
    g0[1] = lds;                                          // lds_addr
    g0[2] = (unsigned int)(ga & 0xFFFFFFFFu);             // global_addr[31:0]
    g0[3] = (unsigned int)((ga >> 32) & 0x1FFFFFFu) | (2u << 30);  // addr[56:32], type=2
    v8i g1;
    g1[0] = (int)(2u << 16);                 // data_size=4B, no multicast/flags
    g1[1] = (int)(8192u << 16);              // tensor_dim0[15:0] << 16
    g1[2] = (int)(1u << 16);                 // tensor_dim0[31:16]=0 | tensor_dim1=1
    g1[3] = (int)(8192u << 16);              // tensor_dim1[31:16]=0 | tile_dim0=8192
    g1[4] = 1;                               // tile_dim1=1, tile_dim2=0 (unused)
    g1[5] = 8192;                            // tensor_dim0_stride[31:0]
    g1[6] = 0;                               // stride0 hi | tensor_dim1_stride lo
    g1[7] = 0;
    v4i g2;
    g2[0] = 1;                               // tensor_dim2=1 (degenerate)
    g2[1] = 1;                               // tensor_dim3=1
    g2[2] = 0;                               // tensor_dim2_stride lo
    g2[3] = 0;                               // stride hi | tile_dim3=0
    v4i g3 = {0, 0, 0, 0};
    v8i g4 = {0, 0, 0, 0, 0, 0, 0, 0};       // clang-23 6-arg form: extra group, zero
    __builtin_amdgcn_tensor_load_to_lds(g0, g1, g2, g3, g4, 0);
    __builtin_amdgcn_s_wait_tensorcnt(0);
  }
  __syncthreads();

  const int m = lane & 15, half = lane >> 4, base = half * 2;
  const int i0 = (blockIdx.x * 8 + wave) * 16;

  v2f a[16];   // full K=64 of the A stripe: K = kk*4 + base + {0,1}
#pragma unroll
  for (int kk = 0; kk < 16; ++kk)
    a[kk] = *(const v2f*)(x + (size_t)(i0 + m) * INC + kk * 4 + base);

#pragma unroll 1
  for (int j = 0; j < 8; ++j) {
    const float bn = bias[j * 16 + m];   // bias depends only on column -> all rows
    v8f acc;
#pragma unroll
    for (int r = 0; r < 8; ++r) acc[r] = bn;
#pragma unroll
    for (int kk = 0; kk < 16; ++kk) {
      const int krow = kk * 4 + base;
      v2f bf;
      bf[0] = sW[krow * OUTC + j * 16 + m];
      bf[1] = sW[(krow + 1) * OUTC + j * 16 + m];
      acc = __builtin_amdgcn_wmma_f32_16x16x4_f32(false, a[kk], false, bf,
                                                  (short)0, acc, false, false);
    }
#pragma unroll
    for (int r = 0; r < 8; ++r)
      h[(size_t)(i0 + r + half * 8) * OUTC + j * 16 + m] = acc[r];
  }
}

// ---------------------------------------------------------------------------
// 4) BN batch statistics: per-channel sum / sumsq, atomicAdd into stats.
// ---------------------------------------------------------------------------
__global__ __launch_bounds__(256) void bn_stats_kernel(const float* __restrict__ h,
                                                       float* __restrict__ stats) {
  __shared__ float ls[256], ls2[256];
  const int tid = threadIdx.x;
  const int c = tid & 127, sub = tid >> 7;
  const int rows = NPTS / 128;          // gridDim.x == 128
  const int r0 = blockIdx.x * rows;
  float s = 0.0f, s2 = 0.0f;
  for (int r = r0 + sub; r < r0 + rows; r += 2) {
    const float v = h[(size_t)r * OUTC + c];
    s += v; s2 += v * v;
  }
  ls[tid] = s; ls2[tid] = s2;
  __syncthreads();
  if (sub == 0) {
    atomicAdd(&stats[c],       ls[tid]  + ls[tid + 128]);
    atomicAdd(&stats[128 + c], ls2[tid] + ls2[tid + 128]);
  }
}

__global__ __launch_bounds__(128) void bn_finalize_kernel(float* __restrict__ stats,
                                                          const float* __restrict__ gamma,
                                                          const float* __restrict__ beta) {
  const int c = threadIdx.x;
  const float mean = stats[c] * (1.0f / NPTS);
  float var = stats[128 + c] * (1.0f / NPTS) - mean * mean;
  var = fmaxf(var, 0.0f);
  const float sc = gamma[c] * rsqrtf(var + 1e-5f);
  stats[256 + c] = sc;
  stats[384 + c] = beta[c] - mean * sc;
}

// 5) feat = relu(h*scale + shift) in place, float4 traffic
__global__ __launch_bounds__(256) void bn_relu_kernel(float* __restrict__ h,
                                                      const float* __restrict__ stats) {
  const size_t i4 = (size_t)blockIdx.x * 256 + threadIdx.x;
  const int c0 = (int)((i4 * 4) & 127);
  v4f v = ((const v4f*)h)[i4];
#pragma unroll
  for (int j = 0; j < 4; ++j) {
    const float sc = stats[256 + c0 + j];
    const float sh = stats[384 + c0 + j];
    v[j] = fmaxf(v[j] * sc + sh, 0.0f);
  }
  ((v4f*)h)[i4] = v;
}

// ---------------------------------------------------------------------------
// 6) kNN via WMMA distance tiles. d_eff = -2 q.p + |p|^2 (row-constant |q|^2
//    dropped; ordering per query preserved). A=[qx,qy,qz,1], B=[-2p, |p|^2].
//    Wave owns 32 queries (2 A-frags); C tiles transposed through wave-private
//    LDS; per-lane register top-16 (replace-worst, unsorted — consumer is max).
// ---------------------------------------------------------------------------
__global__ __launch_bounds__(256) void knn_wmma_kernel(const float* __restrict__ pos,
                                                       const float* __restrict__ subpos,
                                                       int* __restrict__ nn) {
  __shared__ float tiles[8][512];
  const int tid = threadIdx.x;
  const int wave = tid >> 5, lane = tid & 31;
  const int m = lane & 15, half = lane >> 4;
  const int q0 = (blockIdx.x * 8 + wave) * 32;
  float* tl = tiles[wave];

  const int qa = q0 + m, qb = q0 + 16 + m;
  const float ax0 = subpos[qa * 3 + 0], ax1 = subpos[qa * 3 + 1], ax2 = subpos[qa * 3 + 2];
  const float bx0 = subpos[qb * 3 + 0], bx1 = subpos[qb * 3 + 1], bx2 = subpos[qb * 3 + 2];
  v2f aLo, aHi;
  aLo[0] = half ? ax2 : ax0;  aLo[1] = half ? 1.0f : ax1;
  aHi[0] = half ? bx2 : bx0;  aHi[1] = half ? 1.0f : bx1;

  float bestD[KNN]; int bestI[KNN];
#pragma unroll
  for (int s = 0; s < KNN; ++s) { bestD[s] = 3.0e38f; bestI[s] = 0; }
  float worst = 3.0e38f; int wslot = 0;

  for (int p0 = 0; p0 < NPTS; p0 += 16) {
    const int pn = p0 + m;
    const float X = pos[pn * 3 + 0], Y = pos[pn * 3 + 1], Z = pos[pn * 3 + 2];
    __builtin_prefetch(pos + (size_t)(pn + 16) * 3, 0, 1);   // next tile's point
    const float pp = X * X + Y * Y + Z * Z;
    v2f bf;
    bf[0] = half ? (-2.0f * Z) : (-2.0f * X);
    bf[1] = half ? pp          : (-2.0f * Y);

    v8f c0 = {}, c1 = {};
    c0 = __builtin_amdgcn_wmma_f32_16x16x4_f32(false, aLo, false, bf, (short)0, c0, false, false);
    c1 = __builtin_amdgcn_wmma_f32_16x16x4_f32(false, aHi, false, bf, (short)0, c1, false, false);

#pragma unroll
    for (int r = 0; r < 8; ++r) tl[(r + half * 8) * 16 + m] = c0[r];
#pragma unroll
    for (int r = 0; r < 8; ++r) tl[256 + (r + half * 8) * 16 + m] = c1[r];
    asm volatile("s_wait_dscnt 0" ::: "memory");
    __builtin_amdgcn_wave_barrier();

    const float* row = tl + half * 256 + m * 16;   // lane owns query q0+lane
#pragma unroll
    for (int n = 0; n < 16; ++n) {
      const float d = row[n];
      if (d < worst) {
        const int gi = p0 + n;
#pragma unroll
        for (int s = 0; s < KNN; ++s)
          if (s == wslot) { bestD[s] = d; bestI[s] = gi; }
        worst = bestD[0]; wslot = 0;
#pragma unroll
        for (int s = 1; s < KNN; ++s)
          if (bestD[s] > worst) { worst = bestD[s]; wslot = s; }
      }
    }
    __builtin_amdgcn_wave_barrier();
    asm volatile("" ::: "memory");
  }

  const int q = q0 + lane;
#pragma unroll
  for (int s = 0; s < KNN; ++s) nn[q * KNN + s] = bestI[s];
}

// ---------------------------------------------------------------------------
// 7) out[m,c] = max_k feat[nn[m,k], c]
// ---------------------------------------------------------------------------
__global__ __launch_bounds__(128) void gather_max_kernel(const float* __restrict__ feat,
                                                         const int* __restrict__ nn,
                                                         float* __restrict__ out) {
  __shared__ int ln[KNN];
  const int c = threadIdx.x;
  if (c < KNN) ln[c] = nn[(size_t)blockIdx.x * KNN + c];
  __syncthreads();
  float mv = -3.0e38f;
#pragma unroll
  for (int k = 0; k < KNN; ++k)
    mv = fmaxf(mv, feat[(size_t)ln[k] * OUTC + c]);
  out[(size_t)blockIdx.x * OUTC + c] = mv;
}

// ---------------------------------------------------------------------------
extern "C" void kernel_launch(void* const* d_in, const int* in_sizes, int n_in,
                              void* d_out, int out_size, void* d_ws, size_t ws_size,
                              hipStream_t stream) {
  (void)in_sizes; (void)n_in; (void)out_size; (void)ws_size;
  const float* x     = (const float*)d_in[0];
  const float* pos   = (const float*)d_in[1];
  const int*   batch = (const int*)d_in[2];
  const float* W     = (const float*)d_in[3];
  const float* bias  = (const float*)d_in[4];
  const float* gamma = (const float*)d_in[5];
  const float* beta  = (const float*)d_in[6];
  float* out = (float*)d_out;

  float* h     = (float*)d_ws;                    // [N, OUTC] f32 (16 MB)
  float* stats = h + (size_t)NPTS * OUTC;         // 512 f32
  int*   idc   = (int*)(stats + 512);             // [M]
  int*   nn    = idc + MPTS;                      // [M, KNN]

  hipLaunchKernelGGL(init_stats_kernel, dim3(1), dim3(512), 0, stream, stats);
  hipLaunchKernelGGL(fps_kernel, dim3(1), dim3(1024), 0, stream, pos, idc);
  hipLaunchKernelGGL(subgather_kernel, dim3(MPTS / 256), dim3(256), 0, stream,
                     pos, batch, idc, out);
  hipLaunchKernelGGL(gemm_wmma_kernel, dim3(NPTS / 16 / 8), dim3(256), 0, stream,
                     x, W, bias, h);
  hipLaunchKernelGGL(bn_stats_kernel, dim3(128), dim3(256), 0, stream, h, stats);
  hipLaunchKernelGGL(bn_finalize_kernel, dim3(1), dim3(128), 0, stream, stats, gamma, beta);
  hipLaunchKernelGGL(bn_relu_kernel, dim3((NPTS * OUTC) / (4 * 256)), dim3(256), 0, stream,
                     h, stats);
  const float* subposc = out + (size_t)MPTS * OUTC;
  hipLaunchKernelGGL(knn_wmma_kernel, dim3(MPTS / 32 / 8), dim3(256), 0, stream,
                     pos, subposc, nn);
  hipLaunchKernelGGL(gather_max_kernel, dim3(MPTS), dim3(128), 0, stream, h, nn, out);
}